// N_pair_loss_58007828300256
// MI455X (gfx1250) — compile-verified
//
#include <hip/hip_runtime.h>
#include <math.h>

// N-pair loss on MI455X (gfx1250, wave32).
// Memory-bound streaming kernel: 256 MiB read once -> ~11.5us floor at 23.3 TB/s.
// - Non-temporal b128 loads (single-use data; don't rinse the 192MB L2)
// - One fused pass: exp(1-x), per-row partials wave-reduced via
//   v_wmma_f32_16x16x4_f32 against an all-ones B matrix, per-thread column
//   accumulators flushed with fire-and-forget global_atomic_add_f32.
// - Hot loop has NO cross-lane shuffle: after the WMMA, lane 0 holds
//   sum(lanes 0-7,16-23) and lane 16 holds sum(lanes 8-15,24-31); the two
//   half-sums are stored to LDS and folded in the final block reduce,
//   eliminating the ds_bpermute + s_wait_dscnt stall.
// - Diagonal handled analytically in the epilogue (no branch in the hot loop):
//   s_i = rowsum_i + colsum_i - 2*exp(1-d_ii) + 2*e

#define NMAT 8192
#define ROWS_PER_BLOCK 32
#define THREADS 256
#define CHUNKS 8              // 8192 cols = 256 threads * 8 chunks * 4 floats

typedef float v2f __attribute__((ext_vector_type(2)));
typedef float v4f __attribute__((ext_vector_type(4)));
typedef float v8f __attribute__((ext_vector_type(8)));

// V_WMMA_F32_16X16X4_F32 lane reduction core.
// A (16x4 f32, documented layout): lanes 0-15 hold {K0,K1}, lanes 16-31 {K2,K3};
// per-lane value goes in K0/K2, K1/K3 zeroed. B = all-ones (4x16) is
// layout-independent. D[m,n] = x[lane m] + x[lane m+16]; lane l's 8 D VGPRs hold
// rows v + 8*(l/16). Summing them gives per-half partial sums:
//   lanes 0-15:  sum(lanes 0-7)  + sum(lanes 16-23)
//   lanes 16-31: sum(lanes 8-15) + sum(lanes 24-31)
// so halfsum(lane0) + halfsum(lane16) == full 32-lane sum.
__device__ __forceinline__ float wmma_half_sum(float x) {
#if __has_builtin(__builtin_amdgcn_wmma_f32_16x16x4_f32)
  v2f a; a[0] = x;    a[1] = 0.0f;
  v2f b; b[0] = 1.0f; b[1] = 1.0f;
  v8f c = {};
  c = __builtin_amdgcn_wmma_f32_16x16x4_f32(false, a, false, b, (short)0, c, false, false);
  return ((c[0] + c[1]) + (c[2] + c[3])) + ((c[4] + c[5]) + (c[6] + c[7]));
#else
  float t = x;
  t += __shfl_xor(t, 1, 32);
  t += __shfl_xor(t, 2, 32);
  t += __shfl_xor(t, 4, 32);
  t += __shfl_xor(t, 8, 32);
  return 0.5f * t;   // unreachable fallback shape; half-sum semantics approximated
#endif
}

// Full wave32 sum (cold paths only): half-sum + one xor-16 combine.
__device__ __forceinline__ float wave_sum32(float x) {
  float t = wmma_half_sum(x);
  t += __shfl_xor(t, 16, 32);
  return t;
}

__global__ void npair_zero_kernel(float* __restrict__ p, int n) {
  int i = blockIdx.x * blockDim.x + threadIdx.x;
  if (i < n) p[i] = 0.0f;
}

// Pass 1: E = exp(1 - dis); rowsum (exclusive per block, direct store) and
// colsum (shared across blocks, atomic add).
__global__ __launch_bounds__(THREADS)
void npair_pass1_kernel(const float* __restrict__ dis,
                        float* __restrict__ rowsum,
                        float* __restrict__ colsum) {
  __shared__ float lds_row[ROWS_PER_BLOCK * 16];  // [row][wave*2 + half]
  const int tid  = threadIdx.x;
  const int wave = tid >> 5;
  const int lane = tid & 31;
  const int r0   = blockIdx.x * ROWS_PER_BLOCK;

  float colAcc[CHUNKS][4];
#pragma unroll
  for (int k = 0; k < CHUNKS; ++k)
#pragma unroll
    for (int j = 0; j < 4; ++j) colAcc[k][j] = 0.0f;

  for (int r = 0; r < ROWS_PER_BLOCK; ++r) {
    const float* row = dis + (size_t)(r0 + r) * NMAT;
    if (r + 1 < ROWS_PER_BLOCK) {
      // global_prefetch_b8 of the next row's first chunk for this thread
      __builtin_prefetch(row + NMAT + 4 * tid, 0, 1);
    }
    float rp = 0.0f;
#pragma unroll
    for (int k = 0; k < CHUNKS; ++k) {
      const v4f v = __builtin_nontemporal_load(
          (const v4f*)(row + 4 * (tid + THREADS * k)));
      float e0 = __expf(1.0f - v[0]);
      float e1 = __expf(1.0f - v[1]);
      float e2 = __expf(1.0f - v[2]);
      float e3 = __expf(1.0f - v[3]);
      rp += (e0 + e1) + (e2 + e3);
      colAcc[k][0] += e0;
      colAcc[k][1] += e1;
      colAcc[k][2] += e2;
      colAcc[k][3] += e3;
    }
    // No divergence up to here: EXEC is all-ones for the WMMA.
    const float hs = wmma_half_sum(rp);
    // Lanes 0 and 16 together cover all 32 lanes exactly once; store both
    // half-sums, fold in the final reduce. No cross-lane shuffle in hot loop.
    if ((lane & 15) == 0) lds_row[r * 16 + wave * 2 + (lane >> 4)] = hs;
  }
  __syncthreads();

  if (tid < ROWS_PER_BLOCK) {
    float s = 0.0f;
#pragma unroll
    for (int w = 0; w < 16; ++w) s += lds_row[tid * 16 + w];
    rowsum[r0 + tid] = s;
  }

  // Flush column partials: each column is owned by exactly one thread per block,
  // so this is 8192 lane-atomics per block to 32 KB of L2 (non-returning f32 add).
#pragma unroll
  for (int k = 0; k < CHUNKS; ++k) {
    const int c = 4 * (tid + THREADS * k);
    atomicAdd(&colsum[c + 0], colAcc[k][0]);
    atomicAdd(&colsum[c + 1], colAcc[k][1]);
    atomicAdd(&colsum[c + 2], colAcc[k][2]);
    atomicAdd(&colsum[c + 3], colAcc[k][3]);
  }
}

// Pass 2: s_i = rowsum_i + colsum_i + 2e - 2*exp(1 - d_ii);
// loss = sum_i (log(s_i) + d_ii)^2 / (2N). Single block; WMMA wave reductions.
__global__ __launch_bounds__(1024)
void npair_pass2_kernel(const float* __restrict__ dis,
                        const float* __restrict__ rowsum,
                        const float* __restrict__ colsum,
                        float* __restrict__ out) {
  __shared__ float lds[32];
  const int tid = threadIdx.x;
  const float twoE = 5.43656365691809f;  // 2*e

  float acc = 0.0f;
  // N is a multiple of blockDim: all 1024 threads run exactly 8 iterations (no divergence)
  for (int i = tid; i < NMAT; i += 1024) {
    const float d = dis[(size_t)i * NMAT + i];
    const float s = rowsum[i] + colsum[i] + twoE - 2.0f * __expf(1.0f - d);
    const float l = __logf(s) + d;
    acc += l * l;
  }
  const float wsum = wave_sum32(acc);
  if ((tid & 31) == 0) lds[tid >> 5] = wsum;
  __syncthreads();
  if (tid < 32) {                 // wave 0 only, EXEC all-ones within it
    const float tot = wave_sum32(lds[tid]);
    if (tid == 0) out[0] = tot / (2.0f * (float)NMAT);
  }
}

extern "C" void kernel_launch(void* const* d_in, const int* in_sizes, int n_in,
                              void* d_out, int out_size, void* d_ws, size_t ws_size,
                              hipStream_t stream) {
  (void)in_sizes; (void)n_in; (void)out_size; (void)ws_size;
  const float* dis = (const float*)d_in[0];
  float* out    = (float*)d_out;
  float* rowsum = (float*)d_ws;          // [N] floats
  float* colsum = rowsum + NMAT;         // [N] floats (atomic accumulators)

  // ws is poisoned by the harness and never re-initialized between replays:
  // zero the atomic accumulators every call.
  npair_zero_kernel<<<NMAT / 256, 256, 0, stream>>>(colsum, NMAT);
  npair_pass1_kernel<<<NMAT / ROWS_PER_BLOCK, THREADS, 0, stream>>>(dis, rowsum, colsum);
  npair_pass2_kernel<<<1, 1024, 0, stream>>>(dis, rowsum, colsum, out);
}